// FilterDetections_83966610636955
// MI455X (gfx1250) — compile-verified
//
#include <hip/hip_runtime.h>

// ---------------------------------------------------------------------------
// FilterDetections (per-class greedy NMS + global top-k) for MI455X (gfx1250)
//
// Inputs (setup_inputs order):
//   d_in[0] boxes          [8, 50000, 4]  f32
//   d_in[1] classification [8, 50000, 16] f32
//   d_in[2] centerness     [8, 50000, 1]  f32
// Output (concatenated flat, return order):
//   out_boxes  [8,300,4] f32   (offset 0)
//   out_scores [8,300]   f32   (offset 9600)
//   out_labels [8,300]   i32   (offset 12000, int bits)
// ---------------------------------------------------------------------------

#define NB       8
#define NN       50000
#define NC       16
#define MAXDET   300
#define CAP      (NC * MAXDET)      /* 4800 kept candidates max per batch */
#define THREADS  1024               /* 32 wave32 waves -> fills one WGP   */
#define KSTRIPE  49                 /* ceil(NN / THREADS)                 */
#define SCORE_TH 0.05f
#define NMS_TH   0.6f

struct Cand { float s; int idx; };

// wave32 butterfly argmax; ties -> lower index (matches jnp.argmax / top_k)
__device__ __forceinline__ void wave_argmax(float& v, int& i) {
#pragma unroll
  for (int m = 16; m >= 1; m >>= 1) {
    float ov = __shfl_xor(v, m, 32);
    int   oi = __shfl_xor(i, m, 32);
    if (ov > v || (ov == v && oi < i)) { v = ov; i = oi; }
  }
}

// ---------------------------------------------------------------------------
// Kernel 1: one block per (batch, class). Greedy NMS over 50000 anchors.
// Scores live in per-thread registers (49-wide stripe, fully unrolled).
// ---------------------------------------------------------------------------
__global__ __launch_bounds__(THREADS)
void nms_kernel(const float4* __restrict__ boxes,   // [NB*NN]
                const float*  __restrict__ cls,     // [NB*NN*NC]
                const float*  __restrict__ cent,    // [NB*NN]
                Cand*         __restrict__ cand,    // [NB*CAP]
                int*          __restrict__ counters)// [NB]
{
  __shared__ float rv[32];
  __shared__ int   ri[32];
  __shared__ float g_bv;
  __shared__ int   g_bi;

  const int task = blockIdx.x;
  const int b    = task / NC;
  const int c    = task % NC;
  const int tid  = threadIdx.x;
  const int lane = tid & 31;
  const int wid  = tid >> 5;
  const float4* __restrict__ bx = boxes + (size_t)b * NN;

  // ---- initial scores: combined = sqrt(cls*cent) if cls > 0.05 else -1 ----
  float sreg[KSTRIPE];
  float mv = -2.0f;      // running stripe max
  int   mi = NN;         // sentinel index
#pragma unroll
  for (int k = 0; k < KSTRIPE; ++k) {
    const int j = tid + k * THREADS;
    float s = -1.0f;
    if (j < NN) {
      const float cl = cls[((size_t)b * NN + j) * NC + c];
      const float ce = cent[(size_t)b * NN + j];
      const float comb = sqrtf(cl * ce);
      s = (cl > SCORE_TH) ? comb : -1.0f;
      if (s > mv) { mv = s; mi = j; }   // ascending j keeps first-index tie
    }
    sreg[k] = s;
  }

  // ---- greedy NMS: up to 300 picks ----
  for (int iter = 0; iter < MAXDET; ++iter) {
    // block-wide argmax of (mv, mi)
    float v = mv; int i = mi;
    wave_argmax(v, i);
    if (lane == 0) { rv[wid] = v; ri[wid] = i; }
    __syncthreads();
    if (wid == 0) {
      v = rv[lane]; i = ri[lane];      // 32 wave leaders -> one wave
      wave_argmax(v, i);
      if (lane == 0) { g_bv = v; g_bi = i; }
    }
    __syncthreads();
    const float bv = g_bv;
    const int   bi = g_bi;
    if (bv <= 0.0f) break;             // reference: keep = s[i] > 0

    if (tid == 0) {
      const int k = atomicAdd(&counters[b], 1);
      Cand cd; cd.s = bv; cd.idx = c * NN + bi;
      cand[(size_t)b * CAP + k] = cd;
    }

    const float4 bb    = bx[bi];       // broadcast read (same address)
    const float  barea = (bb.z - bb.x) * (bb.w - bb.y);

    // fused: apply suppression + recompute stripe argmax in one sweep
    mv = -2.0f; mi = NN;
#pragma unroll
    for (int k = 0; k < KSTRIPE; ++k) {
      const int j = tid + k * THREADS;
      // gfx1250 prefetch of upcoming stripe tile (speculative, in-bounds)
      if (j + 4 * THREADS < NN) __builtin_prefetch(bx + j + 4 * THREADS, 0, 1);
      const float sj = sreg[k];
      if (j < NN && sj > 0.0f) {
        if (j == bi) {
          sreg[k] = -1.0f;             // .at[i].set(-1)
        } else {
          const float4 pb  = bx[j];
          const float  ix1 = fmaxf(bb.x, pb.x);
          const float  iy1 = fmaxf(bb.y, pb.y);
          const float  ix2 = fminf(bb.z, pb.z);
          const float  iy2 = fminf(bb.w, pb.w);
          const float  inter = fmaxf(ix2 - ix1, 0.0f) * fmaxf(iy2 - iy1, 0.0f);
          const float  parea = (pb.z - pb.x) * (pb.w - pb.y);
          const float  iou   = inter / (barea + parea - inter + 1e-8f);
          if (iou > NMS_TH) sreg[k] = -1.0f;
          else if (sj > mv) { mv = sj; mi = j; }
        }
      }
    }
    // no barrier needed: stripes are thread-private; shared-scratch reuse is
    // protected by the two __syncthreads() inside the reduction above.
  }
}

// ---------------------------------------------------------------------------
// Kernel 2: one block per batch. Top-300 of kept candidates by
// (score desc, flat index asc) — exact jax.lax.top_k tie order.
// ---------------------------------------------------------------------------
__global__ __launch_bounds__(256)
void topk_kernel(const float4* __restrict__ boxes,    // [NB*NN]
                 const Cand*   __restrict__ cand,     // [NB*CAP]
                 const int*    __restrict__ counters, // [NB]
                 float*        __restrict__ out)
{
  __shared__ float sc[CAP];
  __shared__ int   si[CAP];
  __shared__ float rv[8];
  __shared__ int   ri[8];
  __shared__ int   rp[8];
  __shared__ float g_v;
  __shared__ int   g_i;
  __shared__ int   g_p;

  const int b    = blockIdx.x;
  const int tid  = threadIdx.x;
  const int lane = tid & 31;
  const int wid  = tid >> 5;

  int count = counters[b];
  if (count > CAP) count = CAP;

  for (int k = tid; k < count; k += 256) {
    const Cand cd = cand[(size_t)b * CAP + k];
    sc[k] = cd.s; si[k] = cd.idx;
  }
  __syncthreads();

  float* out_boxes  = out;                              // 9600 floats
  float* out_scores = out + NB * MAXDET * 4;            // 2400 floats
  int*   out_labels = (int*)(out + NB * MAXDET * 5);    // 2400 ints

  for (int r = 0; r < MAXDET; ++r) {
    // argmax over remaining candidates: (score desc, idx asc), carry position
    float v = -2.0f; int i = 0x7fffffff; int p = -1;
    for (int k = tid; k < count; k += 256) {
      const float s = sc[k];
      const int idx = si[k];
      if (s > v || (s == v && idx < i)) { v = s; i = idx; p = k; }
    }
#pragma unroll
    for (int m = 16; m >= 1; m >>= 1) {
      const float ov = __shfl_xor(v, m, 32);
      const int   oi = __shfl_xor(i, m, 32);
      const int   op = __shfl_xor(p, m, 32);
      if (ov > v || (ov == v && oi < i)) { v = ov; i = oi; p = op; }
    }
    if (lane == 0) { rv[wid] = v; ri[wid] = i; rp[wid] = p; }
    __syncthreads();
    if (tid == 0) {
      float bvv = rv[0]; int bii = ri[0]; int bpp = rp[0];
      for (int w = 1; w < 8; ++w) {
        if (rv[w] > bvv || (rv[w] == bvv && ri[w] < bii)) {
          bvv = rv[w]; bii = ri[w]; bpp = rp[w];
        }
      }
      g_v = bvv; g_i = bii; g_p = bpp;
    }
    __syncthreads();
    const float bv = g_v;
    const int   bi = g_i;
    const int   bp = g_p;

    if (tid == 0) {
      if (bv > 0.0f) {                       // valid = top_s > 0
        sc[bp] = -2.0f;                      // consume
        const int label  = bi / NN;
        const int anchor = bi - label * NN;
        ((float4*)out_boxes)[b * MAXDET + r] = boxes[(size_t)b * NN + anchor];
        out_scores[b * MAXDET + r] = bv;
        out_labels[b * MAXDET + r] = label;
      } else {
        ((float4*)out_boxes)[b * MAXDET + r] = make_float4(-1.f, -1.f, -1.f, -1.f);
        out_scores[b * MAXDET + r] = -1.0f;
        out_labels[b * MAXDET + r] = -1;
      }
    }
    __syncthreads();                         // sc[bp] visible to next sweep
  }
}

__global__ void init_counters_kernel(int* __restrict__ counters) {
  if (threadIdx.x < NB) counters[threadIdx.x] = 0;
}

// ---------------------------------------------------------------------------
extern "C" void kernel_launch(void* const* d_in, const int* in_sizes, int n_in,
                              void* d_out, int out_size, void* d_ws, size_t ws_size,
                              hipStream_t stream) {
  (void)in_sizes; (void)n_in; (void)out_size; (void)ws_size;

  const float4* boxes = (const float4*)d_in[0];
  const float*  cls   = (const float*)d_in[1];
  const float*  cent  = (const float*)d_in[2];
  float*        out   = (float*)d_out;

  int*  counters = (int*)d_ws;                         // 8 ints
  Cand* cand     = (Cand*)((char*)d_ws + 256);         // 8 * 4800 * 8B

  init_counters_kernel<<<1, 32, 0, stream>>>(counters);
  nms_kernel<<<NB * NC, THREADS, 0, stream>>>(boxes, cls, cent, cand, counters);
  topk_kernel<<<NB, 256, 0, stream>>>(boxes, cand, counters, out);
}